// MultiGridAgentEncoder_87857851007176
// MI455X (gfx1250) — compile-verified
//
#include <hip/hip_runtime.h>
#include <hip/hip_bf16.h>

typedef float v2f __attribute__((ext_vector_type(2)));
typedef float v8f __attribute__((ext_vector_type(8)));

#define MT     64    // rows per block
#define KP     128   // padded K (117 -> 128)
#define NT     64    // output cols per pass
#define NPASS  4     // 256 / NT
#define KREAL  117
#define NOUT   256
#define NAG    16

// Swizzled A-fragment address for X[m][k] within a 64-row tile:
//   16x4 f32 A layout: lane = (m&15) + 16*((k>>1)&1), reg = (k&1)
//   fragment block = (k/4)*numMsub + (m/16), 64 floats per block
__device__ __forceinline__ int xIdx(int m, int k) {
    return ((k >> 2) * 4 + (m >> 4)) * 64 + (k & 1) * 32 + ((k >> 1) & 1) * 16 + (m & 15);
}
// Swizzled B-fragment address for W[k][nLocal] within a 64-col chunk:
//   4x16 f32 B layout: lane = (n&15) + 16*((k>>1)&1), reg = (k&1)
__device__ __forceinline__ int wIdx(int k, int nL) {
    return ((k >> 2) * 4 + (nL >> 4)) * 64 + (k & 1) * 32 + ((k >> 1) & 1) * 16 + (nL & 15);
}

__launch_bounds__(256)
__global__ void multigrid_encoder_wmma_kernel(
    const float* __restrict__ qpos, const float* __restrict__ qdir,
    const float* __restrict__ qabl, const float* __restrict__ qcar,
    const float* __restrict__ qsta,
    const float* __restrict__ apos, const float* __restrict__ adir,
    const float* __restrict__ aabl, const float* __restrict__ acar,
    const float* __restrict__ asta,
    const int*   __restrict__ colors,
    const float* __restrict__ W,    const float* __restrict__ bias,
    float* __restrict__ out)
{
    __shared__ float Xs[MT * KP];   // 32 KB: A fragments, pre-swizzled
    __shared__ float Ws[KP * NT];   // 32 KB: B fragments, pre-swizzled

    const int tid  = threadIdx.x;
    const int lane = tid & 31;
    const int wave = tid >> 5;
    const int rowBase = blockIdx.x * MT;

    // ---- zero X tile (padding k=117..127 and empty agent slots must be 0) ----
    for (int i = tid; i < MT * KP; i += 256) Xs[i] = 0.0f;
    __syncthreads();

    // ---- fused gather: 4 threads cooperate on each row ----
    {
        const int r   = tid >> 2;          // local row 0..63
        const int sub = tid & 3;           // which quarter of the agents
        const int rg  = rowBase + r;       // global row

        if (sub == 0) {  // query features -> k = 0..12
            Xs[xIdx(r, 0)]  = qpos[rg * 2 + 0];
            Xs[xIdx(r, 1)]  = qpos[rg * 2 + 1];
            Xs[xIdx(r, 2)]  = qdir[rg * 4 + 0];
            Xs[xIdx(r, 3)]  = qdir[rg * 4 + 1];
            Xs[xIdx(r, 4)]  = qdir[rg * 4 + 2];
            Xs[xIdx(r, 5)]  = qdir[rg * 4 + 3];
            Xs[xIdx(r, 6)]  = qabl[rg * 2 + 0];
            Xs[xIdx(r, 7)]  = qabl[rg * 2 + 1];
            Xs[xIdx(r, 8)]  = qcar[rg * 2 + 0];
            Xs[xIdx(r, 9)]  = qcar[rg * 2 + 1];
            Xs[xIdx(r, 10)] = qsta[rg * 3 + 0];
            Xs[xIdx(r, 11)] = qsta[rg * 3 + 1];
            Xs[xIdx(r, 12)] = qsta[rg * 3 + 2];
        }

        // sequential rank scan == reference cumsum semantics
        // COLOR_ORDER = ['grey'(idx 5) -> offset 13, 'yellow'(idx 4) -> offset 65]
        int gcnt = 0, ycnt = 0;
        const int nlo = sub * 4, nhi = nlo + 4;
        for (int n = 0; n < NAG; ++n) {
            const int c = colors[rg * NAG + n];
            int base = -1;
            if (c == 5)      { if (gcnt < 4) base = 13 + gcnt * 13; ++gcnt; }
            else if (c == 4) { if (ycnt < 4) base = 65 + ycnt * 13; ++ycnt; }
            if (base >= 0 && n >= nlo && n < nhi) {
                const int a = rg * NAG + n;
                Xs[xIdx(r, base + 0)]  = apos[a * 2 + 0];
                Xs[xIdx(r, base + 1)]  = apos[a * 2 + 1];
                Xs[xIdx(r, base + 2)]  = adir[a * 4 + 0];
                Xs[xIdx(r, base + 3)]  = adir[a * 4 + 1];
                Xs[xIdx(r, base + 4)]  = adir[a * 4 + 2];
                Xs[xIdx(r, base + 5)]  = adir[a * 4 + 3];
                Xs[xIdx(r, base + 6)]  = aabl[a * 2 + 0];
                Xs[xIdx(r, base + 7)]  = aabl[a * 2 + 1];
                Xs[xIdx(r, base + 8)]  = acar[a * 2 + 0];
                Xs[xIdx(r, base + 9)]  = acar[a * 2 + 1];
                Xs[xIdx(r, base + 10)] = asta[a * 3 + 0];
                Xs[xIdx(r, base + 11)] = asta[a * 3 + 1];
                Xs[xIdx(r, base + 12)] = asta[a * 3 + 2];
            }
        }
    }

    // wave -> (M-subtile, pair of N-tiles)
    const int mSub  = wave & 3;    // rows  mSub*16 .. +15
    const int nPair = wave >> 2;   // n-tiles {2*nPair, 2*nPair+1} of this pass

    for (int p = 0; p < NPASS; ++p) {
        __syncthreads();   // X ready (p==0) / previous pass done reading Ws (p>0)

        // ---- stage W column-chunk into B-fragment-swizzled LDS ----
        const int nBaseG = p * NT;
        for (int e = tid; e < KP * NT; e += 256) {
            const int k  = e >> 6;
            const int nL = e & 63;
            const float v = (k < KREAL) ? W[k * NOUT + nBaseG + nL] : 0.0f;
            Ws[wIdx(k, nL)] = v;
        }
        __syncthreads();

        v8f acc0 = {};
        v8f acc1 = {};
        #pragma unroll 4
        for (int k4 = 0; k4 < KP / 4; ++k4) {
            const int aOff = (k4 * 4 + mSub) * 64 + lane;
            v2f A; A.x = Xs[aOff]; A.y = Xs[aOff + 32];

            const int b0Off = (k4 * 4 + nPair * 2 + 0) * 64 + lane;
            v2f B0; B0.x = Ws[b0Off]; B0.y = Ws[b0Off + 32];
            acc0 = __builtin_amdgcn_wmma_f32_16x16x4_f32(
                false, A, false, B0, (short)0, acc0, false, false);

            const int b1Off = (k4 * 4 + nPair * 2 + 1) * 64 + lane;
            v2f B1; B1.x = Ws[b1Off]; B1.y = Ws[b1Off + 32];
            acc1 = __builtin_amdgcn_wmma_f32_16x16x4_f32(
                false, A, false, B1, (short)0, acc1, false, false);
        }

        // ---- epilogue: bias + ReLU + store (C/D layout: m = v + 8*(lane>>4)) ----
        const int m0   = 8 * (lane >> 4);
        const int nIn  = lane & 15;
        const int col0 = nBaseG + (nPair * 2 + 0) * 16 + nIn;
        const int col1 = col0 + 16;
        const float bb0 = bias[col0];
        const float bb1 = bias[col1];
        #pragma unroll
        for (int v = 0; v < 8; ++v) {
            const int row = rowBase + mSub * 16 + m0 + v;
            float o0 = acc0[v] + bb0; o0 = o0 > 0.0f ? o0 : 0.0f;
            float o1 = acc1[v] + bb1; o1 = o1 > 0.0f ? o1 : 0.0f;
            out[row * NOUT + col0] = o0;
            out[row * NOUT + col1] = o1;
        }
    }
}

extern "C" void kernel_launch(void* const* d_in, const int* in_sizes, int n_in,
                              void* d_out, int out_size, void* d_ws, size_t ws_size,
                              hipStream_t stream) {
    const float* qpos   = (const float*)d_in[0];
    const float* qdir   = (const float*)d_in[1];
    const float* qabl   = (const float*)d_in[2];
    const float* qcar   = (const float*)d_in[3];
    const float* qsta   = (const float*)d_in[4];
    const float* apos   = (const float*)d_in[5];
    const float* adir   = (const float*)d_in[6];
    const float* aabl   = (const float*)d_in[7];
    const float* acar   = (const float*)d_in[8];
    const float* asta   = (const float*)d_in[9];
    const int*   colors = (const int*)  d_in[10];
    const float* W      = (const float*)d_in[11];
    const float* bias   = (const float*)d_in[12];
    float* out = (float*)d_out;

    const int Brows = in_sizes[0] / 2;           // query_position is (B, 2)
    const int grid  = (Brows + MT - 1) / MT;     // 131072/64 = 2048 blocks

    multigrid_encoder_wmma_kernel<<<grid, 256, 0, stream>>>(
        qpos, qdir, qabl, qcar, qsta,
        apos, adir, aabl, acar, asta,
        colors, W, bias, out);
}